// gmLL_38328288150154
// MI455X (gfx1250) — compile-verified
//
#include <hip/hip_runtime.h>

// ---------------------------------------------------------------------------
// Problem constants
// ---------------------------------------------------------------------------
#define Dd   99          // data_dim
#define K2   9801        // Dd*Dd
#define NBAS 9800        // Dd*Dd - 1 basis matrices
#define N3   970299      // Dd*K2 (size of t_re / t_im / s_im)
#define KT   49          // k-tiles for basis streaming
#define KLEN 200         // k per tile (49*200 == 9800)
#define BATCH 65536

typedef float v2f __attribute__((ext_vector_type(2)));
typedef float v8f __attribute__((ext_vector_type(8)));

// Workspace layout (float offsets)
#define OFF_PART 0u                 // 2 * KT * K2 = 960498
#define OFF_CRE  960498u
#define OFF_CIM  970299u
#define OFF_TRE  980100u
#define OFF_TIM  1950399u
#define OFF_SIM  2920698u
#define OFF_A1   3890997u
#define OFF_B1   3900798u
#define OFF_B2   3910599u
#define OFF_WT   3920400u           // 100 x 112 padded W^T
#define OFF_BIAS 3931600u           // 99

// ---------------------------------------------------------------------------
// Stage 1: partial sums of  acc[ij] = sum_k basis[k,ij] * Bloch[k]
// 768 MB streamed with non-temporal loads; fully coalesced (threads span ij).
// grid (39, KT, 2) x 256.  z=0 -> basis_re, z=1 -> basis_im.
// ---------------------------------------------------------------------------
__global__ __launch_bounds__(256) void k_basis_partial(
    const float* __restrict__ basis_re, const float* __restrict__ basis_im,
    const float* __restrict__ Bloch, float* __restrict__ part) {
  __shared__ float bl[KLEN];
  const int tid = threadIdx.x;
  const int kb  = blockIdx.y * KLEN;
  const float* basis = (blockIdx.z == 0) ? basis_re : basis_im;
  float* po = part + (size_t)blockIdx.z * ((size_t)KT * K2);
  if (tid < KLEN) bl[tid] = Bloch[kb + tid];
  __syncthreads();
  const int ij = blockIdx.x * 256 + tid;
  if (ij >= K2) return;
  float acc = 0.0f;
  const float* p = basis + (size_t)kb * K2 + ij;
  for (int k = 0; k < KLEN; ++k) {
    acc += __builtin_nontemporal_load(p) * bl[k];   // stream, don't pollute L2
    p += K2;
  }
  po[(size_t)blockIdx.y * K2 + ij] = acc;
}

// ---------------------------------------------------------------------------
// Stage 2: fold partials, apply scale = Tr^2/D, add identity for c_re.
// ---------------------------------------------------------------------------
__global__ __launch_bounds__(256) void k_finalize_c(
    const float* __restrict__ part, const float* __restrict__ Tr,
    float* __restrict__ c_re, float* __restrict__ c_im) {
  const int ij = blockIdx.x * 256 + threadIdx.x;
  if (ij >= K2) return;
  const float t = Tr[0];
  const float scale = t * t / (float)Dd;
  float ar = 0.0f, ai = 0.0f;
  for (int kt = 0; kt < KT; ++kt) {
    ar += part[(size_t)kt * K2 + ij];
    ai += part[(size_t)(KT + kt) * K2 + ij];
  }
  const int i = ij / Dd, j = ij % Dd;
  c_re[ij] = scale * (ar + (i == j ? 1.0f : 0.0f));
  c_im[ij] = scale * ai;
}

// ---------------------------------------------------------------------------
// t_re[n,j,k] = sum_i f[n,i,k] c_re[i,j]      (z=0)
// t_im[n,j,k] = sum_i d[n,i,k] c_im[i,j]      (z=1)
// s_im[n,j,k] = sum_i f[n,i,k] c_im[i,j]      (z=2)   (== u_im)
// One thread per output element; f/d/c_* are L2-resident.
// ---------------------------------------------------------------------------
__global__ __launch_bounds__(256) void k_contract3(
    const float* __restrict__ f, const float* __restrict__ dten,
    const float* __restrict__ c_re, const float* __restrict__ c_im,
    float* __restrict__ t_re, float* __restrict__ t_im, float* __restrict__ s_im) {
  const int idx = blockIdx.x * 256 + threadIdx.x;
  if (idx >= N3) return;
  const int z = blockIdx.y;
  const int n = idx / K2;
  const int rem = idx % K2;
  const int j = rem / Dd;
  const int k = rem % Dd;
  const float* T = (z == 1) ? dten : f;
  const float* C = (z == 0) ? c_re : c_im;
  float acc = 0.0f;
  const float* tp = T + (size_t)n * K2 + k;   // stride Dd over i
  const float* cp = C + j;                    // stride Dd over i
  for (int i = 0; i < Dd; ++i) acc += tp[(size_t)i * Dd] * cp[(size_t)i * Dd];
  float* O = (z == 0) ? t_re : (z == 1) ? t_im : s_im;
  O[idx] = acc;
}

// ---------------------------------------------------------------------------
// A1[m,n] = <f[m,:,:],   t_re[n,:,:]>   (and A2 = A1^T algebraically)
// B1[m,n] = <f[m,:,:],   t_im[n,:,:]>
// B2[m,n] = <s_im[m,:,:], d[n,:,:]>
// One block per (m,n), fixed-order tree reduction (deterministic).
// ---------------------------------------------------------------------------
__global__ __launch_bounds__(256) void k_pairdot(
    const float* __restrict__ f, const float* __restrict__ dten,
    const float* __restrict__ t_re, const float* __restrict__ t_im,
    const float* __restrict__ s_im,
    float* __restrict__ A1, float* __restrict__ B1, float* __restrict__ B2) {
  __shared__ float red[256];
  const int mn = blockIdx.x;
  const int z  = blockIdx.y;
  const int m = mn / Dd, n = mn % Dd;
  const float *X, *Y;
  if (z == 0)      { X = f    + (size_t)m * K2; Y = t_re + (size_t)n * K2; }
  else if (z == 1) { X = f    + (size_t)m * K2; Y = t_im + (size_t)n * K2; }
  else             { X = s_im + (size_t)m * K2; Y = dten + (size_t)n * K2; }
  float acc = 0.0f;
  for (int e = threadIdx.x; e < K2; e += 256) acc += X[e] * Y[e];
  red[threadIdx.x] = acc;
  __syncthreads();
  for (int s = 128; s > 0; s >>= 1) {
    if (threadIdx.x < s) red[threadIdx.x] += red[threadIdx.x + s];
    __syncthreads();
  }
  if (threadIdx.x == 0) {
    float* O = (z == 0) ? A1 : (z == 1) ? B1 : B2;
    O[mn] = red[0];
  }
}

// ---------------------------------------------------------------------------
// Build padded W^T (100 x 112):
//   Weff[m,n] = 1.01*(m==n) + 0.01*( -2*(A1[m,n]+A1[n,m]+B1[m,n]) + 2*B2[m,n]
//                                    + 8*sum_k f[k,m,n]*omega[k] )
// Stored as Wt[n*112 + m]; pads (n>=99 or m>=99) are zero.
// ---------------------------------------------------------------------------
__global__ __launch_bounds__(256) void k_build_W(
    const float* __restrict__ f, const float* __restrict__ omega,
    const float* __restrict__ A1, const float* __restrict__ B1,
    const float* __restrict__ B2, float* __restrict__ Wt) {
  const int idx = blockIdx.x * 256 + threadIdx.x;
  if (idx >= 100 * 112) return;
  const int n = idx / 112, m = idx % 112;
  float v = 0.0f;
  if (m < Dd && n < Dd) {
    float h = 0.0f;
    for (int k = 0; k < Dd; ++k) h += f[(size_t)k * K2 + m * Dd + n] * omega[k];
    const int mn = m * Dd + n, nm = n * Dd + m;
    const float Mmn = -2.0f * (A1[mn] + A1[nm] + B1[mn]) + 2.0f * B2[mn] + 8.0f * h;
    v = 0.01f * Mmn + ((m == n) ? 1.01f : 0.0f);
  }
  Wt[idx] = v;
}

// bias[m] = 0.01 * tr_id[m] = -0.08 * sum_{i,j} f[i,m,j] * c_im[i,j]
__global__ __launch_bounds__(256) void k_bias(
    const float* __restrict__ f, const float* __restrict__ c_im,
    float* __restrict__ bias) {
  __shared__ float red[256];
  const int m = blockIdx.x;
  float acc = 0.0f;
  for (int e = threadIdx.x; e < K2; e += 256) {
    const int i = e / Dd, j = e % Dd;
    acc += f[(size_t)i * K2 + m * Dd + j] * c_im[e];
  }
  red[threadIdx.x] = acc;
  __syncthreads();
  for (int s = 128; s > 0; s >>= 1) {
    if (threadIdx.x < s) red[threadIdx.x] += red[threadIdx.x + s];
    __syncthreads();
  }
  if (threadIdx.x == 0) bias[m] = -0.08f * red[0];
}

// ---------------------------------------------------------------------------
// Batched apply with fp32 WMMA: out[b,m] = sum_n Wt[n,m]*x[b,n] + bias[m]
//   M = 16 batch rows / wave, N = 7 tiles of 16 (99->112), K = 4 per step
//   (99 -> 25 steps, padded col 99 == 0).
// A frag (16x4 f32, ISA 7.12.2): lane l -> M=l&15, K={0,1} (l<16) / {2,3}
//   in the two VGPRs. B frag (4x16) mirrors A. C/D: VGPR r -> M=r+8*(l>=16),
//   N=l&15.
// x tile staged in LDS (128x100 f32, conflict-free: 100r mod 64 is a
// permutation over r=0..15); Weff (44 KB) read straight from L2.
// ---------------------------------------------------------------------------
__global__ __launch_bounds__(256) void k_gemm_wmma(
    const float* __restrict__ x, const float* __restrict__ Wt,
    const float* __restrict__ bias, float* __restrict__ out) {
  __shared__ float xs[128 * 100];
  const int tid = threadIdx.x;
  const long rowBase = (long)blockIdx.x * 128;

  for (int e = tid; e < 128 * 100; e += 256) {
    const int r = e / 100, c = e % 100;
    xs[e] = (c < Dd) ? x[(rowBase + r) * Dd + c] : 0.0f;   // coalesced stage + pad
  }
  __syncthreads();

  const int wave = tid >> 5;
  const int lane = tid & 31;
  const int ml   = lane & 15;
  const int krow = (lane < 16) ? 0 : 2;
  const int wr   = wave * 16;          // this wave's 16 batch rows in LDS

  v8f acc[7];
#pragma unroll
  for (int t = 0; t < 7; ++t) acc[t] = (v8f){0, 0, 0, 0, 0, 0, 0, 0};

  for (int kk = 0; kk < 25; ++kk) {
    const int nb = kk * 4;
    v2f A;
    A.x = xs[(wr + ml) * 100 + nb + krow];
    A.y = xs[(wr + ml) * 100 + nb + krow + 1];
#pragma unroll
    for (int t = 0; t < 7; ++t) {
      v2f Bf;
      Bf.x = Wt[(nb + krow) * 112 + t * 16 + ml];
      Bf.y = Wt[(nb + krow + 1) * 112 + t * 16 + ml];
      acc[t] = __builtin_amdgcn_wmma_f32_16x16x4_f32(
          /*neg_a=*/false, A, /*neg_b=*/false, Bf,
          /*c_mod=*/(short)0, acc[t], /*reuse_a=*/false, /*reuse_b=*/false);
    }
  }

  const int mrow = (lane < 16) ? 0 : 8;
#pragma unroll
  for (int t = 0; t < 7; ++t) {
    const int mg = t * 16 + ml;
    if (mg < Dd) {
      const float bv = bias[mg];
#pragma unroll
      for (int r = 0; r < 8; ++r) {
        const long grow = rowBase + wr + mrow + r;
        out[grow * Dd + mg] = acc[t][r] + bv;
      }
    }
  }
}

// ---------------------------------------------------------------------------
// Launch
// ---------------------------------------------------------------------------
extern "C" void kernel_launch(void* const* d_in, const int* in_sizes, int n_in,
                              void* d_out, int out_size, void* d_ws, size_t ws_size,
                              hipStream_t stream) {
  const float* x        = (const float*)d_in[0];
  const float* Tr       = (const float*)d_in[1];
  const float* Bloch    = (const float*)d_in[2];
  const float* omega    = (const float*)d_in[3];
  const float* basis_re = (const float*)d_in[4];
  const float* basis_im = (const float*)d_in[5];
  const float* f        = (const float*)d_in[6];
  const float* dten     = (const float*)d_in[7];
  float* out = (float*)d_out;
  float* ws  = (float*)d_ws;

  float* part = ws + OFF_PART;
  float* c_re = ws + OFF_CRE;
  float* c_im = ws + OFF_CIM;
  float* t_re = ws + OFF_TRE;
  float* t_im = ws + OFF_TIM;
  float* s_im = ws + OFF_SIM;
  float* A1   = ws + OFF_A1;
  float* B1   = ws + OFF_B1;
  float* B2   = ws + OFF_B2;
  float* Wt   = ws + OFF_WT;
  float* bias = ws + OFF_BIAS;

  k_basis_partial<<<dim3(39, KT, 2), 256, 0, stream>>>(basis_re, basis_im, Bloch, part);
  k_finalize_c<<<39, 256, 0, stream>>>(part, Tr, c_re, c_im);
  k_contract3<<<dim3((N3 + 255) / 256, 3), 256, 0, stream>>>(f, dten, c_re, c_im,
                                                             t_re, t_im, s_im);
  k_pairdot<<<dim3(K2, 3), 256, 0, stream>>>(f, dten, t_re, t_im, s_im, A1, B1, B2);
  k_build_W<<<(100 * 112 + 255) / 256, 256, 0, stream>>>(f, omega, A1, B1, B2, Wt);
  k_bias<<<Dd, 256, 0, stream>>>(f, c_im, bias);
  k_gemm_wmma<<<BATCH / 128, 256, 0, stream>>>(x, Wt, bias, out);
}